// VertexProcessingModule_3925600109027
// MI455X (gfx1250) — compile-verified
//
#include <hip/hip_runtime.h>
#include <math.h>

// Problem constants (match reference)
#define BATCH 4
#define NVERT 512
#define IN_DIM 3
#define E 64
#define NBLK 3
#define QCM_N (3 * E) // 192
#define NEG_SLOPE 0.01f
#define PROB_LD 516 // 512 + 4 pad to decorrelate LDS banks across rows

typedef float v8f __attribute__((ext_vector_type(8)));
typedef float v2f __attribute__((ext_vector_type(2)));

__device__ __forceinline__ float lrelu(float x) {
    return x > 0.f ? x : NEG_SLOPE * x;
}

// ---- WMMA f32 16x16x4 fragment helpers (wave32) ----
// A 16x4 f32: VGPR0 = K=k0 (lanes 0-15) / K=k0+2 (lanes 16-31); VGPR1 = k0+1 / k0+3
__device__ __forceinline__ v2f ld_a(const float* A, int lda, int k0) {
    int lane = threadIdx.x & 31;
    int row = lane & 15;
    int hi = lane >> 4;
    const float* p = A + row * lda + k0 + 2 * hi;
    v2f a;
    a.x = p[0];
    a.y = p[1];
    return a;
}
// B 4x16 f32 (row = K, col = N), symmetric layout
__device__ __forceinline__ v2f ld_b(const float* B, int ldb, int k0) {
    int lane = threadIdx.x & 31;
    int col = lane & 15;
    int hi = lane >> 4;
    v2f b;
    b.x = B[(k0 + 2 * hi) * ldb + col];
    b.y = B[(k0 + 2 * hi + 1) * ldb + col];
    return b;
}
__device__ __forceinline__ v8f wmma4(v8f c, v2f a, v2f b) {
    // (neg_a, A, neg_b, B, c_mod, C, reuse_a, reuse_b)
    return __builtin_amdgcn_wmma_f32_16x16x4_f32(false, a, false, b, (short)0, c,
                                                 false, false);
}

// ---------------- embed: v = vertices @ W_embed + b_embed ----------------
__global__ void embed_kernel(const float* __restrict__ verts,
                             const float* __restrict__ W,
                             const float* __restrict__ bias,
                             float* __restrict__ v, int total) {
    int i = blockIdx.x * blockDim.x + threadIdx.x;
    if (i >= total) return;
    int e = i & (E - 1);
    int bn = i >> 6;
    float acc = bias[e];
#pragma unroll
    for (int d = 0; d < IN_DIM; ++d)
        acc = fmaf(verts[bn * IN_DIM + d], W[d * E + e], acc);
    v[i] = acc;
}

// ------------- mlp: v2 = v + lrelu(v@Wm + bm + v) (M=2048,N=64,K=64) -------------
__global__ __launch_bounds__(128) void mlp_kernel(const float* __restrict__ v,
                                                  const float* __restrict__ Wm,
                                                  const float* __restrict__ bm,
                                                  float* __restrict__ v2) {
    int tile = blockIdx.x * 4 + (threadIdx.x >> 5); // 512 tiles: 128 x 4
    int m0 = (tile >> 2) * 16;
    int n0 = (tile & 3) * 16;
    const float* A = v + (size_t)m0 * E;
    const float* Bp = Wm + n0;
    v8f acc = {};
#pragma unroll
    for (int k0 = 0; k0 < E; k0 += 4)
        acc = wmma4(acc, ld_a(A, E, k0), ld_b(Bp, E, k0));
    int lane = threadIdx.x & 31;
    int col = lane & 15, hi = lane >> 4;
#pragma unroll
    for (int r = 0; r < 8; ++r) {
        int m = m0 + r + 8 * hi;
        int n = n0 + col;
        float base = v[(size_t)m * E + n];
        float x = acc[r] + bm[n] + base;
        v2[(size_t)m * E + n] = base + lrelu(x);
    }
}

// ------------- qcm = v2 @ Wq + bq (M=2048,N=192,K=64) -------------
__global__ __launch_bounds__(128) void qcm_kernel(const float* __restrict__ v2,
                                                  const float* __restrict__ Wq,
                                                  const float* __restrict__ bq,
                                                  float* __restrict__ qcm) {
    int tile = blockIdx.x * 4 + (threadIdx.x >> 5); // 1536 tiles: 128 x 12
    int m0 = (tile / 12) * 16;
    int n0 = (tile % 12) * 16;
    const float* A = v2 + (size_t)m0 * E;
    const float* Bp = Wq + n0;
    v8f acc = {};
#pragma unroll
    for (int k0 = 0; k0 < E; k0 += 4)
        acc = wmma4(acc, ld_a(A, E, k0), ld_b(Bp, QCM_N, k0));
    int lane = threadIdx.x & 31;
    int col = lane & 15, hi = lane >> 4;
#pragma unroll
    for (int r = 0; r < 8; ++r) {
        int m = m0 + r + 8 * hi;
        int n = n0 + col;
        qcm[(size_t)m * QCM_N + n] = acc[r] + bq[n];
    }
}

// ------------- fused attention: qh -> logits -> softmax -> probs@m -------------
// One block per (batch, 16-row q tile). 512 threads = 16 waves.
__global__ __launch_bounds__(512) void attn_kernel(
    const float* __restrict__ qcm,  // (B*NV, 192): [q | c | m]
    const float* __restrict__ mask, // (B, NV)
    const float* __restrict__ Wch, const float* __restrict__ bch,
    const float* __restrict__ Wl, const float* __restrict__ blp,
    const float* __restrict__ temp, float* __restrict__ heads) {
    __shared__ float s_qh[16 * E];         // 4 KB
    __shared__ float s_prob[16 * PROB_LD]; // 33 KB (logits -> probs in place)
    __shared__ float s_wl[E];

    int tid = threadIdx.x;
    int wave = tid >> 5;
    int lane = tid & 31;
    int b = blockIdx.x >> 5; // NV/16 = 32 q-tiles per batch
    int q0 = (blockIdx.x & 31) * 16;

    if (tid < E) s_wl[tid] = Wl[tid];

    // Phase 0: qh(16x64) = q(16x64) @ Wch + bch, via WMMA (waves 0-3)
    const float* qbase = qcm + (size_t)(b * NVERT + q0) * QCM_N;
    if (wave < 4) {
        int n0 = wave * 16;
        v8f acc = {};
#pragma unroll
        for (int k0 = 0; k0 < E; k0 += 4)
            acc = wmma4(acc, ld_a(qbase, QCM_N, k0), ld_b(Wch + n0, E, k0));
        int col = lane & 15, hi = lane >> 4;
#pragma unroll
        for (int r = 0; r < 8; ++r) {
            int m = r + 8 * hi;
            int n = n0 + col;
            s_qh[m * E + n] = acc[r] + bch[n];
        }
    }
    __syncthreads();

    // Phase 1: masked logits. Thread tid owns column v=tid; c row cached in regs
    // and reused across all 16 q rows (16x64 FMA per 64 global loads).
    {
        int vv = tid;
        const float* crow = qcm + (size_t)(b * NVERT + vv) * QCM_N + E;
        float cr[E];
#pragma unroll
        for (int e = 0; e < E; ++e) cr[e] = crow[e];
        float lm = mask[b * NVERT + vv];
        float scale = lm / temp[0];
        float mbias = -99999.f * (1.f - lm);
        float bl0 = blp[0];
        for (int qr = 0; qr < 16; ++qr) {
            float acc = 0.f;
#pragma unroll
            for (int e = 0; e < E; ++e) {
                float z = lrelu(cr[e] + s_qh[qr * E + e]);
                acc = fmaf(z, s_wl[e], acc);
            }
            s_prob[qr * PROB_LD + vv] = (acc + bl0) * scale + mbias;
        }
    }
    __syncthreads();

    // Phase 2: softmax along v per q row; one wave per row (16 rows, 16 waves)
    {
        float* row = s_prob + wave * PROB_LD;
        float mx = -3.0e38f;
        for (int vv = lane; vv < NVERT; vv += 32) mx = fmaxf(mx, row[vv]);
#pragma unroll
        for (int off = 16; off > 0; off >>= 1)
            mx = fmaxf(mx, __shfl_xor(mx, off, 32));
        float sum = 0.f;
        for (int vv = lane; vv < NVERT; vv += 32) {
            float ev = __expf(row[vv] - mx);
            row[vv] = ev;
            sum += ev;
        }
#pragma unroll
        for (int off = 16; off > 0; off >>= 1) sum += __shfl_xor(sum, off, 32);
        float inv = 1.f / sum;
        for (int vv = lane; vv < NVERT; vv += 32)
            row[vv] = row[vv] * inv * mask[b * NVERT + vv];
    }
    __syncthreads();

    // Phase 3: heads(16x64) = probs(16x512) @ m(512x64), WMMA (waves 0-3),
    // K=512 -> 128 wmma per wave. lrelu folded into the store.
    if (wave < 4) {
        int n0 = wave * 16;
        const float* mb = qcm + (size_t)b * NVERT * QCM_N + 2 * E + n0;
        v8f acc = {};
#pragma unroll 4
        for (int k0 = 0; k0 < NVERT; k0 += 4)
            acc = wmma4(acc, ld_a(s_prob, PROB_LD, k0), ld_b(mb, QCM_N, k0));
        int col = lane & 15, hi = lane >> 4;
#pragma unroll
        for (int r = 0; r < 8; ++r) {
            int m = r + 8 * hi;
            int n = n0 + col;
            heads[(size_t)(b * NVERT + q0 + m) * E + n] = lrelu(acc[r]);
        }
    }
}

// ------------- reduce: out = heads_lrelu @ Wr + br + v2 (M=2048,N=64,K=64) -------------
__global__ __launch_bounds__(128) void reduce_kernel(
    const float* __restrict__ heads, const float* __restrict__ Wr,
    const float* __restrict__ br, const float* __restrict__ v2,
    float* __restrict__ out) {
    int tile = blockIdx.x * 4 + (threadIdx.x >> 5);
    int m0 = (tile >> 2) * 16;
    int n0 = (tile & 3) * 16;
    const float* A = heads + (size_t)m0 * E;
    const float* Bp = Wr + n0;
    v8f acc = {};
#pragma unroll
    for (int k0 = 0; k0 < E; k0 += 4)
        acc = wmma4(acc, ld_a(A, E, k0), ld_b(Bp, E, k0));
    int lane = threadIdx.x & 31;
    int col = lane & 15, hi = lane >> 4;
#pragma unroll
    for (int r = 0; r < 8; ++r) {
        int m = m0 + r + 8 * hi;
        int n = n0 + col;
        out[(size_t)m * E + n] = acc[r] + br[n] + v2[(size_t)m * E + n];
    }
}

extern "C" void kernel_launch(void* const* d_in, const int* in_sizes, int n_in,
                              void* d_out, int out_size, void* d_ws,
                              size_t ws_size, hipStream_t stream) {
    const float* vertices = (const float*)d_in[0];
    const float* mask = (const float*)d_in[1];
    const float* W_embed = (const float*)d_in[2];
    const float* b_embed = (const float*)d_in[3];
    const float* W_mlp = (const float*)d_in[4];
    const float* b_mlp = (const float*)d_in[5];
    const float* W_qcm = (const float*)d_in[6];
    const float* b_qcm = (const float*)d_in[7];
    const float* W_ch = (const float*)d_in[8];
    const float* b_ch = (const float*)d_in[9];
    const float* W_logit = (const float*)d_in[10];
    const float* b_logit = (const float*)d_in[11];
    const float* W_red = (const float*)d_in[12];
    const float* b_red = (const float*)d_in[13];
    const float* temp = (const float*)d_in[14];

    float* ws = (float*)d_ws;
    float* v = ws;                          // B*NV*E
    float* v2 = v + BATCH * NVERT * E;      // B*NV*E
    float* qcm = v2 + BATCH * NVERT * E;    // B*NV*192
    float* heads = qcm + BATCH * NVERT * QCM_N; // B*NV*E (lrelu already applied)

    int total = BATCH * NVERT * E;
    embed_kernel<<<(total + 255) / 256, 256, 0, stream>>>(vertices, W_embed,
                                                          b_embed, v, total);

    for (int i = 0; i < NBLK; ++i) {
        mlp_kernel<<<128, 128, 0, stream>>>(v, W_mlp + i * E * E, b_mlp + i * E,
                                            v2);
        qcm_kernel<<<384, 128, 0, stream>>>(v2, W_qcm + i * E * QCM_N,
                                            b_qcm + i * QCM_N, qcm);
        attn_kernel<<<BATCH * NVERT / 16, 512, 0, stream>>>(
            qcm, mask, W_ch + i * E * E, b_ch + i * E, W_logit + i * E,
            b_logit + i, temp + i, heads);
        float* dst = (i == NBLK - 1) ? (float*)d_out : v;
        reduce_kernel<<<128, 128, 0, stream>>>(heads, W_red + i * E * E,
                                               b_red + i * E, v2, dst);
    }
}